// EncoderLayer_22024592294472
// MI455X (gfx1250) — compile-verified
//
#include <hip/hip_runtime.h>
#include <hip/hip_bf16.h>

// -------------------- CDNA5 fp32 WMMA encoder layer --------------------
// GEMMs use V_WMMA_F32_16X16X4_F32 (wave32 fragment layouts per ISA 7.12.2):
//   A frag (16x4): lane L holds (m = L%16, k = 2*(L/16)+v), v in {0,1}
//   B frag (4x16): lane L holds (k = 2*(L/16)+v, n = L%16)
//   C/D   (16x16): lane L, vgpr r holds (m = r + 8*(L/16), n = L%16)
// Global->LDS staging uses CDNA5 async loads (ASYNCcnt) with double buffering.

typedef float v2f __attribute__((ext_vector_type(2)));
typedef float v8f __attribute__((ext_vector_type(8)));
typedef int   v4i __attribute__((ext_vector_type(4)));

#define EMB   1024
#define HEADS 16
#define QKV   64
#define INNER 2816
#define BATCH 4
#define SEQ   2048
#define MTOT  (BATCH * SEQ)   // 8192

#define LSTRIDE 20            // LDS row stride in floats: 80B, 16B-aligned, conflict-free

static __device__ __forceinline__ v8f wmma_f32(v2f a, v2f b, v8f c) {
  return __builtin_amdgcn_wmma_f32_16x16x4_f32(false, a, false, b, (short)0, c,
                                               false, false);
}

// async 16B global -> LDS copy (GLOBAL_LOAD_ASYNC_TO_LDS_B128, ASYNCcnt)
static __device__ __forceinline__ void async_copy16(const float* g, float* l) {
  __builtin_amdgcn_global_load_async_to_lds_b128(
      (__attribute__((address_space(1))) v4i*)(g),
      (__attribute__((address_space(3))) v4i*)(l), 0, 0);
}

static __device__ __forceinline__ void wait_async0() {
  asm volatile("s_wait_asynccnt 0x0" ::: "memory");
}

// C[M,N] = A[M,K] @ W[N,K]^T  (+ residual[M,N] + bias[N])
// grid = (N/128, M/128), block = 256 threads (8 waves)
__global__ __launch_bounds__(256) void gemm_f32_wmma(
    const float* __restrict__ A, const float* __restrict__ W,
    const float* __restrict__ residual, const float* __restrict__ bias,
    float* __restrict__ C, int M, int N, int K) {
  __shared__ __align__(16) float ldsA[2][128 * LSTRIDE];
  __shared__ __align__(16) float ldsB[2][128 * LSTRIDE];

  const int tid  = threadIdx.x;
  const int lane = tid & 31;
  const int wave = tid >> 5;
  const int hi   = lane >> 4;     // 0/1
  const int lo   = lane & 15;
  const int mBase = blockIdx.y * 128;
  const int nBase = blockIdx.x * 128;
  const int wm = (wave >> 2) * 64;   // wave's M offset in block tile
  const int wn = (wave & 3) * 32;    // wave's N offset in block tile

  v8f acc[4][2];
#pragma unroll
  for (int i = 0; i < 4; ++i)
#pragma unroll
    for (int j = 0; j < 2; ++j) acc[i][j] = (v8f)0.0f;

  const int ldRow = tid >> 1;       // 0..127
  const int ldCol = (tid & 1) * 8;  // 0 or 8
  const float* gaBase = A + (size_t)(mBase + ldRow) * K + ldCol;
  const float* gbBase = W + (size_t)(nBase + ldRow) * K + ldCol;
  const int lofs = ldRow * LSTRIDE + ldCol;

  // prologue: fill buffer 0
  async_copy16(gaBase, &ldsA[0][lofs]);
  async_copy16(gaBase + 4, &ldsA[0][lofs + 4]);
  async_copy16(gbBase, &ldsB[0][lofs]);
  async_copy16(gbBase + 4, &ldsB[0][lofs + 4]);
  wait_async0();
  __syncthreads();

  int cur = 0;
  for (int k0 = 0; k0 < K; k0 += 16) {
    // stream next K-slab into the other buffer while computing on this one
    if (k0 + 16 < K) {
      const int nxt = cur ^ 1;
      const float* ga = gaBase + k0 + 16;
      const float* gb = gbBase + k0 + 16;
      async_copy16(ga, &ldsA[nxt][lofs]);
      async_copy16(ga + 4, &ldsA[nxt][lofs + 4]);
      async_copy16(gb, &ldsB[nxt][lofs]);
      async_copy16(gb + 4, &ldsB[nxt][lofs + 4]);
    }

#pragma unroll
    for (int kk = 0; kk < 4; ++kk) {
      const int kc = kk * 4 + hi * 2;
      v2f af[4], bf[2];
#pragma unroll
      for (int i = 0; i < 4; ++i)
        af[i] = *(const v2f*)&ldsA[cur][(wm + i * 16 + lo) * LSTRIDE + kc];
#pragma unroll
      for (int j = 0; j < 2; ++j)
        bf[j] = *(const v2f*)&ldsB[cur][(wn + j * 16 + lo) * LSTRIDE + kc];
#pragma unroll
      for (int i = 0; i < 4; ++i)
#pragma unroll
        for (int j = 0; j < 2; ++j) acc[i][j] = wmma_f32(af[i], bf[j], acc[i][j]);
    }

    wait_async0();
    __syncthreads();
    cur ^= 1;
  }

  // epilogue
#pragma unroll
  for (int i = 0; i < 4; ++i) {
#pragma unroll
    for (int j = 0; j < 2; ++j) {
      const int n = nBase + wn + j * 16 + lo;
      const float b = bias ? bias[n] : 0.0f;
#pragma unroll
      for (int r = 0; r < 8; ++r) {
        const int m = mBase + wm + i * 16 + r + hi * 8;
        float v = acc[i][j][r] + b;
        if (residual) v += residual[(size_t)m * N + n];
        C[(size_t)m * N + n] = v;
      }
    }
  }
}

// Flash attention over Q,K,V stored as [B, N, H*dk] (= [8192, 1024]).
// grid = (64 b*h, 32 row-tiles), block = 128 (4 waves; each wave = 16 rows).
__global__ __launch_bounds__(128) void attention_f32_wmma(
    const float* __restrict__ Q, const float* __restrict__ Kt,
    const float* __restrict__ V, float* __restrict__ Out) {
  __shared__ __align__(16) float ldsP[4][16 * LSTRIDE];

  const int lane = threadIdx.x & 31;
  const int wave = threadIdx.x >> 5;
  const int hi = lane >> 4;
  const int lo = lane & 15;
  const int bh = blockIdx.x;       // b*16 + h
  const int b  = bh >> 4;
  const int h  = bh & 15;
  const int m0 = blockIdx.y * 64 + wave * 16;  // query row tile (within seq)

  const size_t base = (size_t)b * SEQ * EMB + (size_t)h * QKV;

  // Q fragments (pre-scaled by 1/sqrt(dk))
  v2f qf[16];
#pragma unroll
  for (int kk = 0; kk < 16; ++kk) {
    v2f q = *(const v2f*)&Q[base + (size_t)(m0 + lo) * EMB + kk * 4 + hi * 2];
    qf[kk].x = q.x * 0.125f;
    qf[kk].y = q.y * 0.125f;
  }

  v8f o[4];
#pragma unroll
  for (int t = 0; t < 4; ++t) o[t] = (v8f)0.0f;
  float mrun[8], lrun[8];
#pragma unroll
  for (int r = 0; r < 8; ++r) { mrun[r] = -1e30f; lrun[r] = 0.0f; }

  float* lp = ldsP[wave];

  for (int j = 0; j < SEQ / 16; ++j) {
    // S = Qs * K^T   (16x16 score tile)
    const float* kp = &Kt[base + (size_t)(j * 16 + lo) * EMB + hi * 2];
    v8f s = (v8f)0.0f;
#pragma unroll
    for (int kk = 0; kk < 16; ++kk) {
      v2f kf = *(const v2f*)(kp + kk * 4);
      s = wmma_f32(qf[kk], kf, s);
    }

    // online softmax (rows live in 16-lane halves of each C vgpr)
#pragma unroll
    for (int r = 0; r < 8; ++r) {
      float xm = s[r];
#pragma unroll
      for (int off = 1; off < 16; off <<= 1) xm = fmaxf(xm, __shfl_xor(xm, off));
      const float mnew = fmaxf(mrun[r], xm);
      const float sc = __expf(mrun[r] - mnew);
      const float pe = __expf(s[r] - mnew);
      float rs = pe;
#pragma unroll
      for (int off = 1; off < 16; off <<= 1) rs += __shfl_xor(rs, off);
      lrun[r] = lrun[r] * sc + rs;
      mrun[r] = mnew;
#pragma unroll
      for (int t = 0; t < 4; ++t) o[t][r] *= sc;
      s[r] = pe;
    }

    // stage P through per-wave LDS tile: C-layout -> A-layout
#pragma unroll
    for (int r = 0; r < 8; ++r) lp[(r + hi * 8) * LSTRIDE + lo] = s[r];
    __builtin_amdgcn_sched_barrier(0);

    // O += P * V
#pragma unroll
    for (int kk = 0; kk < 4; ++kk) {
      v2f pf = *(const v2f*)&lp[lo * LSTRIDE + kk * 4 + hi * 2];
      const int krow = j * 16 + kk * 4 + hi * 2;
      const float* vp = &V[base + (size_t)krow * EMB + lo];
#pragma unroll
      for (int t = 0; t < 4; ++t) {
        v2f vf;
        vf.x = vp[t * 16];
        vf.y = vp[(size_t)EMB + t * 16];
        o[t] = wmma_f32(pf, vf, o[t]);
      }
    }
  }

  // normalize and store: Out[b, n, h*64 + d]
#pragma unroll
  for (int r = 0; r < 8; ++r) {
    const float inv = 1.0f / lrun[r];
    const int m = m0 + r + hi * 8;
#pragma unroll
    for (int t = 0; t < 4; ++t)
      Out[base + (size_t)m * EMB + t * 16 + lo] = o[t][r] * inv;
  }
}

// out = rmsnorm(A (+Bres), g) over rows of length EMB; grid = MTOT rows.
__global__ __launch_bounds__(256) void rmsnorm_kernel(
    const float* __restrict__ A, const float* __restrict__ Bres,
    const float* __restrict__ g, float* __restrict__ O) {
  __shared__ float red[8];
  const int row = blockIdx.x;
  const int tid = threadIdx.x;
  const int lane = tid & 31;
  const int wave = tid >> 5;
  const float* a = A + (size_t)row * EMB;
  const float* br = Bres ? Bres + (size_t)row * EMB : nullptr;

  float vals[4];
  float ss = 0.0f;
#pragma unroll
  for (int i = 0; i < 4; ++i) {
    const int c = i * 256 + tid;
    float v = a[c];
    if (br) v += br[c];
    vals[i] = v;
    ss += v * v;
  }
#pragma unroll
  for (int off = 1; off < 32; off <<= 1) ss += __shfl_xor(ss, off);
  if (lane == 0) red[wave] = ss;
  __syncthreads();
  float tot = 0.0f;
#pragma unroll
  for (int w = 0; w < 8; ++w) tot += red[w];
  const float inv = rsqrtf(tot * (1.0f / EMB) + 1e-6f);
#pragma unroll
  for (int i = 0; i < 4; ++i) {
    const int c = i * 256 + tid;
    O[(size_t)row * EMB + c] = vals[i] * inv * g[c];
  }
}

// a <- silu(a) * b, elementwise
__global__ __launch_bounds__(256) void silu_gate_kernel(
    float* __restrict__ a, const float* __restrict__ b) {
  const size_t i = (size_t)blockIdx.x * 256 + threadIdx.x;
  const float av = a[i];
  a[i] = av * b[i] / (1.0f + __expf(-av));
}

extern "C" void kernel_launch(void* const* d_in, const int* in_sizes, int n_in,
                              void* d_out, int out_size, void* d_ws,
                              size_t ws_size, hipStream_t stream) {
  const float* x  = (const float*)d_in[0];
  const float* Wq = (const float*)d_in[1];
  const float* Wk = (const float*)d_in[2];
  const float* Wv = (const float*)d_in[3];
  const float* Wc = (const float*)d_in[4];
  const float* bc = (const float*)d_in[5];
  const float* g1 = (const float*)d_in[6];
  const float* g2 = (const float*)d_in[7];
  const float* L1 = (const float*)d_in[8];
  const float* L2 = (const float*)d_in[9];
  const float* L3 = (const float*)d_in[10];
  float* out = (float*)d_out;
  float* ws  = (float*)d_ws;

  const size_t TD = (size_t)MTOT * EMB;    // 8,388,608 floats
  const size_t TI = (size_t)MTOT * INNER;  // 23,068,672 floats
  float* q    = ws;            // later reused as tmp (x + mha)
  float* kb   = ws + TD;       // later reused as h
  float* vb   = ws + 2 * TD;
  float* attn = ws + 3 * TD;   // later reused as ff
  float* abuf = ws + 4 * TD;
  float* bbuf = abuf + TI;
  float* tmp = q;
  float* h   = kb;
  float* ff  = attn;
  // total ws use: (4*TD + 2*TI)*4 = 318,767,104 bytes

  const dim3 blk256(256);
  const dim3 gD(EMB / 128, MTOT / 128);    // N=1024 GEMMs
  const dim3 gI(INNER / 128, MTOT / 128);  // N=2816 GEMMs

  // QKV projections
  gemm_f32_wmma<<<gD, blk256, 0, stream>>>(x, Wq, nullptr, nullptr, q, MTOT, EMB, EMB);
  gemm_f32_wmma<<<gD, blk256, 0, stream>>>(x, Wk, nullptr, nullptr, kb, MTOT, EMB, EMB);
  gemm_f32_wmma<<<gD, blk256, 0, stream>>>(x, Wv, nullptr, nullptr, vb, MTOT, EMB, EMB);

  // attention
  attention_f32_wmma<<<dim3(BATCH * HEADS, SEQ / 64), dim3(128), 0, stream>>>(q, kb, vb, attn);

  // output projection + bias + residual, then RMSNorm -> h
  gemm_f32_wmma<<<gD, blk256, 0, stream>>>(attn, Wc, x, bc, tmp, MTOT, EMB, EMB);
  rmsnorm_kernel<<<MTOT, blk256, 0, stream>>>(tmp, nullptr, g1, h);

  // gated MLP
  gemm_f32_wmma<<<gI, blk256, 0, stream>>>(h, L1, nullptr, nullptr, abuf, MTOT, INNER, EMB);
  gemm_f32_wmma<<<gI, blk256, 0, stream>>>(h, L2, nullptr, nullptr, bbuf, MTOT, INNER, EMB);
  silu_gate_kernel<<<dim3((unsigned)(TI / 256)), blk256, 0, stream>>>(abuf, bbuf);
  gemm_f32_wmma<<<gD, blk256, 0, stream>>>(abuf, L3, nullptr, nullptr, ff, MTOT, EMB, INNER);

  // final residual + RMSNorm -> out
  rmsnorm_kernel<<<MTOT, blk256, 0, stream>>>(h, ff, g2, out);
}